// ArmInternalGraphNet_50672024158728
// MI455X (gfx1250) — compile-verified
//
#include <hip/hip_runtime.h>
#include <hip/hip_bf16.h>

#define N_NODES 50000
#define N_EDGES 600000
#define DIM     128
#define LN_EPS  1e-5f

typedef __attribute__((ext_vector_type(2))) float v2f;
typedef __attribute__((ext_vector_type(4))) float v4f;
typedef __attribute__((ext_vector_type(8))) float v8f;

// ---------------------------------------------------------------------------
// Degree / normalization: deg[i] = 1 (self loop) + #edges with dst==i,
// then dinv[i] = rsqrt(deg[i])  (deg >= 1 always, so the where() is moot).
// ---------------------------------------------------------------------------
__global__ void deg_init_kernel(float* __restrict__ deg) {
    int i = blockIdx.x * blockDim.x + threadIdx.x;
    if (i < N_NODES) deg[i] = 1.0f;
}

__global__ void deg_count_kernel(const int* __restrict__ dst, float* __restrict__ deg) {
    int e = blockIdx.x * blockDim.x + threadIdx.x;
    if (e < N_EDGES) unsafeAtomicAdd(&deg[dst[e]], 1.0f);
}

__global__ void dinv_kernel(float* __restrict__ deg) {
    int i = blockIdx.x * blockDim.x + threadIdx.x;
    if (i < N_NODES) deg[i] = rsqrtf(deg[i]);
}

// ---------------------------------------------------------------------------
// Wt[n][k] = W[k][n]  (128x128). Makes the WMMA B-fragment load contiguous.
// ---------------------------------------------------------------------------
__global__ void transpose128_kernel(const float* __restrict__ W, float* __restrict__ Wt) {
    int idx = blockIdx.x * blockDim.x + threadIdx.x;   // 16384 elements
    int k = idx >> 7;
    int n = idx & 127;
    Wt[(size_t)n * DIM + k] = W[idx];
}

// ---------------------------------------------------------------------------
// H = X @ W  via V_WMMA_F32_16X16X4_F32, W given pre-transposed (Wt[n][k]).
// Block = 256 threads = 8 waves; block owns 16 output rows, wave w owns cols
// [16w,16w+16). The 16x128 X tile (8KB) is staged in LDS once per block so
// the 8 waves don't re-read it from global (A feeds from ds_load_b64).
// A (16x4 f32): lanes 0-15 -> M=lane, VGPR0/1 = K0/K1; lanes 16-31 -> K2/K3.
// B (4x16 f32): lanes 0-15 -> N=lane, VGPR0/1 = K0/K1; lanes 16-31 -> K2/K3.
// C/D (16x16 f32): VGPR v -> row v (lanes 0-15) / v+8 (lanes 16-31).
// Grid exactly covers 50000 rows (3125*16) -> no divergence, EXEC all 1s.
// ---------------------------------------------------------------------------
__global__ void __launch_bounds__(256)
gemm_wmma_kernel(const float* __restrict__ X, const float* __restrict__ Wt,
                 float* __restrict__ H) {
    __shared__ float xs[16 * DIM];                     // 8 KB X tile

    const int tid     = threadIdx.x;
    const int rowBase = blockIdx.x * 16;

    // Cooperative tile load: 2048 floats = 256 threads x 2 x float4 (b128).
    {
        const v4f* __restrict__ s = (const v4f*)(X + (size_t)rowBase * DIM);
        v4f* d = (v4f*)xs;
        d[tid]       = s[tid];
        d[tid + 256] = s[tid + 256];
    }
    __syncthreads();

    const int wave = tid >> 5;
    const int lane = tid & 31;
    const int l15  = lane & 15;
    const int hi   = lane >> 4;                        // 0: K0/K1 half, 1: K2/K3 half
    const int col  = wave * 16 + l15;

    const float* __restrict__ xrow  = xs + l15 * DIM;          // A from LDS
    const float* __restrict__ wtcol = Wt + (size_t)col * DIM;  // B contiguous

    v8f c = {};
#pragma unroll 8
    for (int k0 = 0; k0 < DIM; k0 += 4) {
        const int ka = k0 + hi * 2;
        v2f a = *(const v2f*)(xrow + ka);              // ds_load_b64
        v2f b = *(const v2f*)(wtcol + ka);             // global_load_b64
        c = __builtin_amdgcn_wmma_f32_16x16x4_f32(
                /*neg_a=*/false, a, /*neg_b=*/false, b,
                /*c_mod=*/(short)0, c, /*reuse_a=*/false, /*reuse_b=*/false);
    }

    const int rowOff = hi * 8;
#pragma unroll
    for (int v = 0; v < 8; ++v)
        H[(size_t)(rowBase + rowOff + v) * DIM + col] = c[v];
}

// ---------------------------------------------------------------------------
// out[i] = bias + dinv[i]^2 * h[i]   (self-loop contribution + bias folded in)
// ---------------------------------------------------------------------------
__global__ void agg_init_kernel(const float* __restrict__ h, const float* __restrict__ dinv,
                                const float* __restrict__ bias, float* __restrict__ out) {
    int idx = blockIdx.x * blockDim.x + threadIdx.x;
    int row = idx >> 5;
    int c4  = (idx & 31) * 4;
    if (row < N_NODES) {
        float dv  = dinv[row];
        float nrm = dv * dv;
        v4f hv = *(const v4f*)(h + (size_t)row * DIM + c4);
        v4f bv = *(const v4f*)(bias + c4);
        v4f o  = hv * nrm + bv;
        *(v4f*)(out + (size_t)row * DIM + c4) = o;
    }
}

// ---------------------------------------------------------------------------
// out[dst] += dinv[src]*dinv[dst] * h[src]   — one wave per edge,
// 32 lanes x float4 = 128 cols; hardware global_atomic_add_f32, no CAS.
// ---------------------------------------------------------------------------
__global__ void agg_edge_kernel(const float* __restrict__ h, const int* __restrict__ src,
                                const int* __restrict__ dst, const float* __restrict__ dinv,
                                float* __restrict__ out) {
    int wid  = (blockIdx.x * blockDim.x + threadIdx.x) >> 5;
    int lane = threadIdx.x & 31;
    if (wid >= N_EDGES) return;

    int s = src[wid];
    int d = dst[wid];
    float nrm = dinv[s] * dinv[d];

    float* op = out + (size_t)d * DIM + lane * 4;
    __builtin_prefetch(op, 1, 3);                      // global_prefetch_b8
    v4f hv = *(const v4f*)(h + (size_t)s * DIM + lane * 4);

    unsafeAtomicAdd(op + 0, hv.x * nrm);
    unsafeAtomicAdd(op + 1, hv.y * nrm);
    unsafeAtomicAdd(op + 2, hv.z * nrm);
    unsafeAtomicAdd(op + 3, hv.w * nrm);
}

// ---------------------------------------------------------------------------
// In-place LayerNorm (+ optional ReLU). One wave per row, __shfl_xor wave32
// butterfly reductions for mean and variance.
// ---------------------------------------------------------------------------
__global__ void ln_kernel(float* __restrict__ A, const float* __restrict__ gamma,
                          const float* __restrict__ beta, int do_relu) {
    int wid  = (blockIdx.x * blockDim.x + threadIdx.x) >> 5;
    int lane = threadIdx.x & 31;
    if (wid >= N_NODES) return;

    float* rowp = A + (size_t)wid * DIM + lane * 4;
    v4f v = *(const v4f*)rowp;

    float s = v.x + v.y + v.z + v.w;
#pragma unroll
    for (int m = 16; m >= 1; m >>= 1) s += __shfl_xor(s, m, 32);
    float mu = s * (1.0f / DIM);

    float dx = v.x - mu, dy = v.y - mu, dz = v.z - mu, dw = v.w - mu;
    float q = dx * dx + dy * dy + dz * dz + dw * dw;
#pragma unroll
    for (int m = 16; m >= 1; m >>= 1) q += __shfl_xor(q, m, 32);
    float rstd = rsqrtf(q * (1.0f / DIM) + LN_EPS);

    v4f g  = *(const v4f*)(gamma + lane * 4);
    v4f bt = *(const v4f*)(beta + lane * 4);
    v4f o;
    o.x = dx * rstd * g.x + bt.x;
    o.y = dy * rstd * g.y + bt.y;
    o.z = dz * rstd * g.z + bt.z;
    o.w = dw * rstd * g.w + bt.w;
    if (do_relu) {
        o.x = fmaxf(o.x, 0.0f);
        o.y = fmaxf(o.y, 0.0f);
        o.z = fmaxf(o.z, 0.0f);
        o.w = fmaxf(o.w, 0.0f);
    }
    *(v4f*)rowp = o;
}

// ---------------------------------------------------------------------------
extern "C" void kernel_launch(void* const* d_in, const int* in_sizes, int n_in,
                              void* d_out, int out_size, void* d_ws, size_t ws_size,
                              hipStream_t stream) {
    const float* x   = (const float*)d_in[0];
    const int*   ei  = (const int*)  d_in[1];
    const float* W1  = (const float*)d_in[2];
    const float* b1  = (const float*)d_in[3];
    const float* g1  = (const float*)d_in[4];
    const float* bt1 = (const float*)d_in[5];
    const float* W2  = (const float*)d_in[6];
    const float* b2  = (const float*)d_in[7];
    const float* g2  = (const float*)d_in[8];
    const float* bt2 = (const float*)d_in[9];
    float* out = (float*)d_out;

    const int* src = ei;            // edge_index[0]
    const int* dst = ei + N_EDGES;  // edge_index[1]

    float* dinv = (float*)d_ws;                         // 50000 floats (padded)
    float* bufA = dinv + 50048;                         // 50000*128 floats
    float* bufB = bufA + (size_t)N_NODES * DIM;         // 50000*128 floats
    float* Wt1  = bufB + (size_t)N_NODES * DIM;         // 128*128 floats
    float* Wt2  = Wt1  + DIM * DIM;                     // 128*128 floats

    const dim3 blk(256);
    const int gN   = (N_NODES + 255) / 256;             // 196
    const int gE   = (N_EDGES + 255) / 256;             // 2344
    const int gRow = N_NODES * 32 / 256;                // 6250 (exact)
    const int gEdW = N_EDGES * 32 / 256;                // 75000 (exact)
    const int gGem = N_NODES / 16;                      // 3125 (exact)
    const int gT   = DIM * DIM / 256;                   // 64

    // normalization + weight transposes (independent, cheap)
    deg_init_kernel    <<<gN, blk, 0, stream>>>(dinv);
    deg_count_kernel   <<<gE, blk, 0, stream>>>(dst, dinv);
    dinv_kernel        <<<gN, blk, 0, stream>>>(dinv);
    transpose128_kernel<<<gT, blk, 0, stream>>>(W1, Wt1);
    transpose128_kernel<<<gT, blk, 0, stream>>>(W2, Wt2);

    // layer 1: h = X@W1 ; agg = b1 + dinv^2*h + scatter ; LN + ReLU
    gemm_wmma_kernel<<<gGem, blk, 0, stream>>>(x, Wt1, bufA);
    agg_init_kernel <<<gRow, blk, 0, stream>>>(bufA, dinv, b1, bufB);
    agg_edge_kernel <<<gEdW, blk, 0, stream>>>(bufA, src, dst, dinv, bufB);
    ln_kernel       <<<gRow, blk, 0, stream>>>(bufB, g1, bt1, 1);

    // layer 2: h2 = a1@W2 ; agg -> d_out ; LN (no relu), in place
    gemm_wmma_kernel<<<gGem, blk, 0, stream>>>(bufB, Wt2, bufA);
    agg_init_kernel <<<gRow, blk, 0, stream>>>(bufA, dinv, b2, out);
    agg_edge_kernel <<<gEdW, blk, 0, stream>>>(bufA, src, dst, dinv, out);
    ln_kernel       <<<gRow, blk, 0, stream>>>(out, g2, bt2, 0);
}